// TTFNetLoss_19645180412522
// MI455X (gfx1250) — compile-verified
//
#include <hip/hip_runtime.h>
#include <hip/hip_bf16.h>

// TTFNet loss for MI455X (gfx1250, wave32).
// No GEMM in this workload -> no WMMA path; this is a bandwidth-bound fp32
// elementwise/reduction pipeline (~88 MB of head tensors, ~4us HBM floor at
// 23.3 TB/s). Target generation exploits separability of the per-object
// Gaussian and per-class object lists so the focal-loss kernel does ~O(1)
// target work per element. Object-parameter staging into LDS uses the
// gfx1250 async global->LDS DMA path (GLOBAL_LOAD_ASYNC_TO_LDS_B128).

#define B_     16
#define NC_    80
#define H_     128
#define W_     128
#define HW_    (H_ * W_)
#define MAXO_  100
#define OP_STRIDE 16   // floats per object record

// objP record layout: 0 cx, 1 cy, 2 wr, 3 hr, 4 2sx^2, 5 2sy^2,
//                     6 valid, 7 gsum, 8..11 x1 y1 x2 y2, 12 cls
// acc layout: 0 pos_loss, 1 neg_loss, 2 num_pos, 3 giou_num, 4 w_sum

#if __has_builtin(__builtin_amdgcn_global_load_async_to_lds_b128)
#define HAVE_ASYNC_LDS 1
typedef __attribute__((__vector_size__(4 * sizeof(int)))) int v4i_t;
typedef __attribute__((address_space(1))) v4i_t as1_v4i;
typedef __attribute__((address_space(3))) v4i_t as3_v4i;
#endif

static __device__ __forceinline__ void async_copy16(const float* gsrc, float* ldst) {
#ifdef HAVE_ASYNC_LDS
  __builtin_amdgcn_global_load_async_to_lds_b128((as1_v4i*)gsrc, (as3_v4i*)ldst, 0, 0);
#else
  *(float4*)ldst = *(const float4*)gsrc;
#endif
}

static __device__ __forceinline__ void wait_async_copies() {
#ifdef HAVE_ASYNC_LDS
#if __has_builtin(__builtin_amdgcn_s_wait_asynccnt)
  __builtin_amdgcn_s_wait_asynccnt(0);
#else
  asm volatile("s_wait_asynccnt 0" ::: "memory");
#endif
#endif
}

__global__ void zero_acc_kernel(double* __restrict__ acc) {
  if (threadIdx.x < 8) acc[threadIdx.x] = 0.0;
}

__global__ void prep_kernel(const float* __restrict__ ann,
                            float* __restrict__ objP,
                            int* __restrict__ clsCnt,
                            int* __restrict__ clsList) {
  const int b = blockIdx.x;
  const int t = threadIdx.x;
  __shared__ float sb[MAXO_][5];
  __shared__ float key[MAXO_];
  __shared__ int   perm[MAXO_];

  for (int i = t; i < MAXO_ * 5; i += blockDim.x)
    sb[i / 5][i % 5] = ann[b * MAXO_ * 5 + i];
  for (int c = t; c < NC_; c += blockDim.x)
    clsCnt[b * NC_ + c] = 0;
  __syncthreads();

  if (t < MAXO_) {
    float x1 = sb[t][0], y1 = sb[t][1], x2 = sb[t][2], y2 = sb[t][3];
    float cls = sb[t][4];
    bool valid = (cls >= 0.0f);
    float area = (y2 - y1 + 1.0f) * (x2 - x1 + 1.0f);
    key[t] = valid ? logf(fmaxf(area, 1e-12f)) : -INFINITY;
  }
  __syncthreads();

  // stable descending rank == argsort(-key) stable ascending
  if (t < MAXO_) {
    float kt = key[t];
    int r = 0;
    for (int j = 0; j < MAXO_; ++j) {
      float kj = key[j];
      if (kj > kt || (kj == kt && j < t)) ++r;
    }
    perm[r] = t;
  }
  __syncthreads();

  if (t < MAXO_) {
    int o = perm[t];  // t = sorted slot
    float x1 = sb[o][0], y1 = sb[o][1], x2 = sb[o][2], y2 = sb[o][3];
    float cls = sb[o][4];
    bool valid = (cls >= 0.0f);

    float cx = truncf((x1 + x2) * 0.5f * 0.25f);   // /DOWN, DOWN=4 exact
    float cy = truncf((y1 + y2) * 0.5f * 0.25f);
    float fx1 = fminf(fmaxf(x1 * 0.25f, 0.0f), (float)(W_ - 1));
    float fx2 = fminf(fmaxf(x2 * 0.25f, 0.0f), (float)(W_ - 1));
    float fy1 = fminf(fmaxf(y1 * 0.25f, 0.0f), (float)(H_ - 1));
    float fy2 = fminf(fmaxf(y2 * 0.25f, 0.0f), (float)(H_ - 1));
    int hr = (int)((fy2 - fy1) * 0.5f * 0.54f);    // trunc toward 0
    int wr = (int)((fx2 - fx1) * 0.5f * 0.54f);
    float sx = (float)(2 * wr + 1) / 6.0f;
    float sy = (float)(2 * hr + 1) / 6.0f;
    float wrf = (float)wr, hrf = (float)hr;
    float n2sx2 = 2.0f * sx * sx;
    float n2sy2 = 2.0f * sy * sy;

    // gsum via separability: sum g = (sum_y gy) * (sum_x gx)
    float sgx = 0.0f, sgy = 0.0f;
    for (int x = 0; x < W_; ++x) {
      float dx = (float)x - cx;
      if (fabsf(dx) <= wrf && valid) sgx += expf(-(dx * dx) / n2sx2);
    }
    for (int y = 0; y < H_; ++y) {
      float dy = (float)y - cy;
      if (fabsf(dy) <= hrf) sgy += expf(-(dy * dy) / n2sy2);
    }
    float gsum = fmaxf(sgy * sgx, 1e-12f);

    float* p = objP + (b * MAXO_ + t) * OP_STRIDE;
    p[0] = cx;  p[1] = cy;  p[2] = wrf; p[3] = hrf;
    p[4] = n2sx2; p[5] = n2sy2;
    p[6] = valid ? 1.0f : 0.0f; p[7] = gsum;
    p[8] = x1;  p[9] = y1;  p[10] = x2; p[11] = y2;
    p[12] = cls; p[13] = 0.0f; p[14] = 0.0f; p[15] = 0.0f;

    if (valid) {
      int c = (int)cls;  // in [0, NC_-1] by construction
      int pos = atomicAdd(&clsCnt[b * NC_ + c], 1);
      clsList[(b * NC_ + c) * MAXO_ + pos] = t;  // sorted-slot index
    }
  }
}

// one block per (b, c, 256-pixel tile); 64 tiles per image plane
__global__ void hm_loss_kernel(const float* __restrict__ heat,
                               const float* __restrict__ objP,
                               const int* __restrict__ clsCnt,
                               const int* __restrict__ clsList,
                               double* __restrict__ acc) {
  const int bid  = blockIdx.x;
  const int tile = bid & 63;
  const int c    = (bid >> 6) % NC_;
  const int b    = bid / (64 * NC_);
  const int tid  = threadIdx.x;

  __shared__ __align__(16) float op[MAXO_][OP_STRIDE];
  __shared__ int s_cnt;
  __shared__ double r0[256], r1[256], r2[256];

  const int pix = tile * 256 + tid;
  const float* hp = heat + (size_t)(b * NC_ + c) * HW_ + pix;
  __builtin_prefetch(hp, 0, 0);  // global_prefetch_b8 on gfx1250

  if (tid == 0) s_cnt = clsCnt[b * NC_ + c];
  __syncthreads();
  const int cnt = s_cnt;
  // gather this class's object records into LDS via async DMA:
  // 4 lanes per record, one b128 each (per-lane scattered addresses are fine)
  for (int i = tid; i < cnt * 4; i += blockDim.x) {
    int k = i >> 2, q = i & 3;
    int s = clsList[(b * NC_ + c) * MAXO_ + k];
    async_copy16(objP + (b * MAXO_ + s) * OP_STRIDE + q * 4, &op[k][q * 4]);
  }
  wait_async_copies();
  __syncthreads();

  const float xf = (float)(pix & (W_ - 1));
  const float yf = (float)(pix >> 7);

  float tgt = 0.0f;  // per-class scatter-max of g (order independent)
  for (int k = 0; k < cnt; ++k) {
    float dx = xf - op[k][0];
    float dy = yf - op[k][1];
    if (fabsf(dx) <= op[k][2] && fabsf(dy) <= op[k][3]) {
      float g = expf(-(dx * dx) / op[k][4]) * expf(-(dy * dy) / op[k][5]);
      tgt = fmaxf(tgt, g);
    }
  }

  float h = *hp;
  float pred = 1.0f / (1.0f + expf(-h));
  double posl = 0.0, negl = 0.0, npos = 0.0;
  if (tgt == 1.0f) {
    float om = 1.0f - pred;
    posl = (double)(-logf(pred) * om * om);  // FOCAL_A = 2
    npos = 1.0;
  } else {  // tgt < 1
    float nw = 1.0f - tgt;
    nw = nw * nw; nw = nw * nw;              // (1-t)^4, FOCAL_B = 4
    negl = (double)(-logf(1.0f - pred) * pred * pred * nw);
  }

  r0[tid] = posl; r1[tid] = negl; r2[tid] = npos;
  __syncthreads();
  for (int s = 128; s > 0; s >>= 1) {
    if (tid < s) {
      r0[tid] += r0[tid + s];
      r1[tid] += r1[tid + s];
      r2[tid] += r2[tid + s];
    }
    __syncthreads();
  }
  if (tid == 0) {
    atomicAdd(&acc[0], r0[0]);
    atomicAdd(&acc[1], r1[0]);
    atomicAdd(&acc[2], r2[0]);
  }
}

// one block per (b, 256-pixel tile)
__global__ void giou_loss_kernel(const float* __restrict__ wh,
                                 const float* __restrict__ objP,
                                 double* __restrict__ acc) {
  const int bid  = blockIdx.x;
  const int tile = bid & 63;
  const int b    = bid >> 6;
  const int tid  = threadIdx.x;

  __shared__ __align__(16) float P[MAXO_ * OP_STRIDE];
  __shared__ double r0[256], r1[256];

  const float* src = objP + (size_t)b * MAXO_ * OP_STRIDE;
  for (int i = tid; i < MAXO_ * OP_STRIDE / 4; i += blockDim.x)
    async_copy16(src + i * 4, &P[i * 4]);
  wait_async_copies();
  __syncthreads();

  const int pix = tile * 256 + tid;
  const float xf = (float)(pix & (W_ - 1));
  const float yf = (float)(pix >> 7);

  // objects are sorted by descending area; owner = last coverer
  int owner = -1;
  float gown = 0.0f;
  for (int s = 0; s < MAXO_; ++s) {
    const float* o = &P[s * OP_STRIDE];
    if (o[6] == 0.0f) continue;
    float dx = xf - o[0], dy = yf - o[1];
    if (fabsf(dx) <= o[2] && fabsf(dy) <= o[3]) {
      float g = expf(-(dx * dx) / o[4]) * expf(-(dy * dy) / o[5]);
      if (g > 0.0f) { owner = s; gown = g; }
    }
  }

  double lw = 0.0, wsum = 0.0;
  if (owner >= 0) {
    const float* o = &P[owner * OP_STRIDE];
    float w = gown / o[7];
    const float* whb = wh + (size_t)b * 4 * HW_ + pix;
    float w0 = whb[0], w1 = whb[HW_], w2 = whb[2 * HW_], w3 = whb[3 * HW_];
    float px1 = xf * 4.0f - w0, py1 = yf * 4.0f - w1;
    float px2 = xf * 4.0f + w2, py2 = yf * 4.0f + w3;
    float tx1 = o[8], ty1 = o[9], tx2 = o[10], ty2 = o[11];

    float ltx = fmaxf(px1, tx1), lty = fmaxf(py1, ty1);
    float rbx = fminf(px2, tx2), rby = fminf(py2, ty2);
    float iw = fmaxf(rbx - ltx + 1.0f, 0.0f);
    float ih = fmaxf(rby - lty + 1.0f, 0.0f);
    float e1x = fminf(px1, tx1), e1y = fminf(py1, ty1);
    float e2x = fmaxf(px2, tx2), e2y = fmaxf(py2, ty2);
    float ew = fmaxf(e2x - e1x + 1.0f, 0.0f);
    float eh = fmaxf(e2y - e1y + 1.0f, 0.0f);
    float ov = iw * ih;
    float ap = (px2 - px1 + 1.0f) * (py2 - py1 + 1.0f);
    float ag = (tx2 - tx1 + 1.0f) * (ty2 - ty1 + 1.0f);
    float u  = ap + ag - ov;
    float iou = ov / u;
    float ea  = ew * eh;
    float giou = iou - (ea - u) / ea;
    lw   = (double)((1.0f - giou) * w);
    wsum = (double)w;
  }

  r0[tid] = lw; r1[tid] = wsum;
  __syncthreads();
  for (int s = 128; s > 0; s >>= 1) {
    if (tid < s) { r0[tid] += r0[tid + s]; r1[tid] += r1[tid + s]; }
    __syncthreads();
  }
  if (tid == 0) {
    atomicAdd(&acc[3], r0[0]);
    atomicAdd(&acc[4], r1[0]);
  }
}

__global__ void finalize_kernel(const double* __restrict__ acc,
                                float* __restrict__ out) {
  double posl = acc[0], negl = acc[1], npos = acc[2];
  double lw = acc[3], wsum = acc[4];
  double hm = (npos > 0.0) ? (posl + negl) / npos : negl;  // HM_W = 1.0
  double whl = (lw / (wsum + 1e-4)) * 0.1;                 // WH_W = 0.1
  out[0] = (float)hm;
  out[1] = (float)whl;
}

extern "C" void kernel_launch(void* const* d_in, const int* in_sizes, int n_in,
                              void* d_out, int out_size, void* d_ws, size_t ws_size,
                              hipStream_t stream) {
  const float* heat = (const float*)d_in[0];  // (16, 80, 128, 128)
  const float* wh   = (const float*)d_in[1];  // (16, 4, 128, 128)
  const float* ann  = (const float*)d_in[2];  // (16, 100, 5)

  char* ws = (char*)d_ws;
  float* objP   = (float*)ws;                                     // 102400 B
  int*   clsCnt = (int*)(ws + 102400);                            //   5120 B
  int*   clsLst = (int*)(ws + 102400 + 5120);                     // 512000 B
  double* acc   = (double*)(ws + 102400 + 5120 + 512000);         //     64 B

  zero_acc_kernel<<<1, 32, 0, stream>>>(acc);
  prep_kernel<<<B_, 128, 0, stream>>>(ann, objP, clsCnt, clsLst);
  hm_loss_kernel<<<B_ * NC_ * 64, 256, 0, stream>>>(heat, objP, clsCnt, clsLst, acc);
  giou_loss_kernel<<<B_ * 64, 256, 0, stream>>>(wh, objP, acc);
  finalize_kernel<<<1, 1, 0, stream>>>(acc, (float*)d_out);
}